// TransitionDown_42520176231040
// MI455X (gfx1250) — compile-verified
//
#include <hip/hip_runtime.h>

#define B_    8
#define N_    8192
#define CIN   128
#define COUT  256
#define NP    2048      // N_/4
#define NS    16
#define KDIM  131
#define KPAD  132
#define AROW  140       // LDS row stride: 560B = 16B aligned, conflict-free cols
#define BN_EPS 1e-5f

typedef float v2f __attribute__((ext_vector_type(2)));
typedef float v8f __attribute__((ext_vector_type(8)));
typedef int   v4i __attribute__((vector_size(4 * sizeof(int))));

#define LDS_AS __attribute__((address_space(3)))
#define GLB_AS __attribute__((address_space(1)))

#if __has_builtin(__builtin_amdgcn_global_load_async_to_lds_b128)
#define HAVE_ASYNC_LDS 1
#else
#define HAVE_ASYNC_LDS 0
#endif

__device__ __forceinline__ void async_wait0() {
#if __has_builtin(__builtin_amdgcn_s_wait_asynccnt)
  __builtin_amdgcn_s_wait_asynccnt(0);
#else
  asm volatile("s_wait_asynccnt 0" ::: "memory");
#endif
}

// ---------------------------------------------------------------------------
// 1) Farthest point sampling: one block per batch, running distance in LDS.
// ---------------------------------------------------------------------------
__global__ __launch_bounds__(512) void fps_kernel(const float* __restrict__ xyz,
                                                  float* __restrict__ new_xyz,
                                                  int* __restrict__ fps_idx) {
  const int b = blockIdx.x;
  const int t = threadIdx.x;
  __shared__ float dist[N_];
  __shared__ float redd[512];
  __shared__ int   redi[512];
  __shared__ int   s_far;
  const float* px = xyz + (size_t)b * N_ * 3;

  for (int i = t; i < N_; i += 512) dist[i] = 1e10f;
  if (t == 0) s_far = 0;
  __syncthreads();

  for (int it = 0; it < NP; ++it) {
    const int far = s_far;
    const float cx = px[far * 3 + 0];
    const float cy = px[far * 3 + 1];
    const float cz = px[far * 3 + 2];
    if (t == 0) {
      fps_idx[b * NP + it] = far;
      new_xyz[(b * NP + it) * 3 + 0] = cx;
      new_xyz[(b * NP + it) * 3 + 1] = cy;
      new_xyz[(b * NP + it) * 3 + 2] = cz;
    }
    float bd = -1.0f;
    int   bi = 0;
    for (int i = t; i < N_; i += 512) {
      const float dx = px[i * 3 + 0] - cx;
      const float dy = px[i * 3 + 1] - cy;
      const float dz = px[i * 3 + 2] - cz;
      const float d  = dx * dx + dy * dy + dz * dz;
      const float nd = fminf(dist[i], d);
      dist[i] = nd;
      if (nd > bd) { bd = nd; bi = i; }
    }
    redd[t] = bd;
    redi[t] = bi;
    __syncthreads();
    for (int s = 256; s > 0; s >>= 1) {
      if (t < s) {
        if (redd[t + s] > redd[t]) { redd[t] = redd[t + s]; redi[t] = redi[t + s]; }
      }
      __syncthreads();
    }
    if (t == 0) s_far = redi[0];
    __syncthreads();
  }
}

// ---------------------------------------------------------------------------
// 2) KNN (top-16): 1 thread per centroid, register-sorted top-16; candidate
//    points staged through LDS in chunks.
// ---------------------------------------------------------------------------
#define KNN_BLOCK 128
#define KNN_CHUNK 1024

__global__ __launch_bounds__(KNN_BLOCK) void knn_kernel(const float* __restrict__ xyz,
                                                        const float* __restrict__ new_xyz,
                                                        int* __restrict__ knn_idx) {
  __shared__ float sx[KNN_CHUNK], sy[KNN_CHUNK], sz[KNN_CHUNK];
  const int g = blockIdx.x * KNN_BLOCK + threadIdx.x;
  const int b = (blockIdx.x * KNN_BLOCK) / NP;
  const float* px = xyz + (size_t)b * N_ * 3;

  const float qx = new_xyz[g * 3 + 0];
  const float qy = new_xyz[g * 3 + 1];
  const float qz = new_xyz[g * 3 + 2];

  float hd[NS];
  int   hi[NS];
#pragma unroll
  for (int m = 0; m < NS; ++m) { hd[m] = 1e30f; hi[m] = 0; }

  for (int base = 0; base < N_; base += KNN_CHUNK) {
    __syncthreads();
    for (int i = threadIdx.x; i < KNN_CHUNK; i += KNN_BLOCK) {
      sx[i] = px[(base + i) * 3 + 0];
      sy[i] = px[(base + i) * 3 + 1];
      sz[i] = px[(base + i) * 3 + 2];
    }
    __syncthreads();
    for (int i = 0; i < KNN_CHUNK; ++i) {
      const float dx = sx[i] - qx;
      const float dy = sy[i] - qy;
      const float dz = sz[i] - qz;
      const float d  = dx * dx + dy * dy + dz * dz;
      if (d < hd[NS - 1]) {
        hd[NS - 1] = d;
        hi[NS - 1] = base + i;
#pragma unroll
        for (int m = NS - 1; m > 0; --m) {
          if (hd[m] < hd[m - 1]) {
            const float td = hd[m]; hd[m] = hd[m - 1]; hd[m - 1] = td;
            const int   ti = hi[m]; hi[m] = hi[m - 1]; hi[m - 1] = ti;
          }
        }
      }
    }
  }
#pragma unroll
  for (int m = 0; m < NS; ++m) knn_idx[g * NS + m] = hi[m];
}

// ---------------------------------------------------------------------------
// 3) BN-stat scratch init / finalize
//    stats layout: [0,256)=sum [256,512)=sumsq [512,768)=mean [768,1024)=invstd
// ---------------------------------------------------------------------------
__global__ void zero_stats(float* __restrict__ stats) {
  const int i = threadIdx.x;
  if (i < 512) stats[i] = 0.0f;
}

__global__ void finalize_stats(float* __restrict__ stats) {
  const int c = threadIdx.x;  // 256 threads
  const float inv_cnt = 1.0f / (float)(B_ * NP * NS);
  const float mean = stats[c] * inv_cnt;
  const float var  = stats[256 + c] * inv_cnt - mean * mean;
  stats[512 + c] = mean;
  stats[768 + c] = rsqrtf(var + BN_EPS);
}

// ---------------------------------------------------------------------------
// 4) Fused gather + WMMA GEMM.
//    K dimension is reordered: cols [0,128) = features, 128..130 = rel xyz,
//    131 = zero pad (GEMM sum over K is order-invariant; B rows permuted to
//    match). Feature rows (512B, 16B-aligned) stream into LDS via
//    GLOBAL_LOAD_ASYNC_TO_LDS_B128. B fragments (33 x v2f per wave) are
//    preloaded into registers once: the hot loop is A-ds-load + wmma only.
//    MODE 0: per-channel sum/sumsq of h.  MODE 1: BN + ReLU + max-over-k.
// ---------------------------------------------------------------------------
template <int MODE>
__global__ __launch_bounds__(256) void gemm_kernel(
    const float* __restrict__ xyz, const float* __restrict__ features,
    const float* __restrict__ W, const float* __restrict__ gamma,
    const float* __restrict__ beta, const float* __restrict__ new_xyz,
    const int* __restrict__ knn_idx, float* __restrict__ stats,
    float* __restrict__ out_features) {
  __shared__ __align__(16) float As[16][AROW];  // 16 x 140 floats
  __shared__ float ssum[128], ssq[128];
  __shared__ int   sid[16];

  const int tid  = threadIdx.x;
  const int lane = tid & 31;
  const int wave = tid >> 5;    // 0..7
  const int half = lane >> 4;   // 0/1 -> K pair and M half
  const int l16  = lane & 15;
  const int cbase = blockIdx.y * 128;
  const int nbase = wave * 16;
  const int cglob = cbase + nbase + l16;   // this lane's output channel
  const int koff  = half * 2;

  // Preload this lane's 33 B fragments (permuted K rows) into registers.
  auto loadW = [&](int l) -> float {
    if (l >= KDIM) return 0.0f;                       // zero pad row
    const int row = (l < CIN) ? (l + 3) : (l - CIN);  // feat rows then xyz rows
    return W[row * COUT + cglob];
  };
  v2f breg[33];
#pragma unroll
  for (int s = 0; s < 33; ++s) {
    const int l0 = 4 * s + koff;
    breg[s].x = loadW(l0);
    breg[s].y = loadW(l0 + 1);
  }

  if (MODE == 0 && tid < 128) { ssum[tid] = 0.0f; ssq[tid] = 0.0f; }

  float mean = 0.0f, inv = 0.0f, ga = 0.0f, be = 0.0f;
  if (MODE == 1) {
    mean = stats[512 + cglob];
    inv  = stats[768 + cglob];
    ga   = gamma[cglob];
    be   = beta[cglob];
  }
  __syncthreads();

  for (int g = blockIdx.x; g < B_ * NP; g += gridDim.x) {
    const int b = g / NP;
    __syncthreads();  // protect As/sid from previous iteration's readers
    if (tid < 16) sid[tid] = knn_idx[g * NS + tid];
    __syncthreads();

    // --- Feature gather: 16 rows x 512B, global -> LDS, 16B chunks. ---
#pragma unroll
    for (int i = 0; i < 2; ++i) {
      const int q  = tid + 256 * i;   // 512 chunks total
      const int r  = q >> 5;          // row 0..15
      const int ch = q & 31;          // 16B chunk within row
      const float* src = features + ((size_t)b * N_ + sid[r]) * CIN + ch * 4;
      float* dst = &As[r][ch * 4];
#if HAVE_ASYNC_LDS
      __builtin_amdgcn_global_load_async_to_lds_b128(
          (GLB_AS v4i*)src, (LDS_AS v4i*)dst, 0, 0);
#else
      *(float4*)dst = *(const float4*)src;
#endif
    }
    // --- Relative xyz (cols 128..130) + zero pad (col 131). ---
    if (tid < 64) {
      const int r = tid >> 2;
      const int j = tid & 3;
      float v = 0.0f;
      if (j < 3) v = xyz[((size_t)b * N_ + sid[r]) * 3 + j] - new_xyz[g * 3 + j];
      As[r][CIN + j] = v;
    }
#if HAVE_ASYNC_LDS
    async_wait0();
#endif
    __syncthreads();

    // --- 16x16 tile: 33 x (A ds-load + WMMA), B already in registers. ---
    v8f acc = {};
#pragma unroll
    for (int s = 0; s < 33; ++s) {
      v2f a;
      a.x = As[l16][4 * s + koff];
      a.y = As[l16][4 * s + koff + 1];
      acc = __builtin_amdgcn_wmma_f32_16x16x4_f32(
          false, a, false, breg[s], (short)0, acc, false, false);
    }

    if (MODE == 0) {
      float s = 0.0f, q = 0.0f;
#pragma unroll
      for (int j = 0; j < 8; ++j) { const float v = acc[j]; s += v; q += v * v; }
      atomicAdd(&ssum[nbase + l16], s);
      atomicAdd(&ssq[nbase + l16], q);
    } else {
      float m = -1e30f;
#pragma unroll
      for (int j = 0; j < 8; ++j) {
        float v = (acc[j] - mean) * inv * ga + be;   // BN affine
        v = fmaxf(v, 0.0f);                          // ReLU
        m = fmaxf(m, v);                             // max over 8 rows
      }
      m = fmaxf(m, __shfl_xor(m, 16, 32));           // merge M halves
      if (half == 0)
        out_features[(size_t)g * COUT + cglob] = m;
    }
  }

  if (MODE == 0) {
    __syncthreads();
    if (tid < 128) {
      atomicAdd(&stats[cbase + tid], ssum[tid]);
      atomicAdd(&stats[256 + cbase + tid], ssq[tid]);
    }
  }
}

// ---------------------------------------------------------------------------
extern "C" void kernel_launch(void* const* d_in, const int* in_sizes, int n_in,
                              void* d_out, int out_size, void* d_ws, size_t ws_size,
                              hipStream_t stream) {
  const float* xyz      = (const float*)d_in[0];
  const float* features = (const float*)d_in[1];
  const float* W        = (const float*)d_in[2];
  const float* gamma    = (const float*)d_in[3];
  const float* beta     = (const float*)d_in[4];

  float* out          = (float*)d_out;
  float* new_xyz      = out;                       // [B, NP, 3]
  float* new_features = out + (size_t)B_ * NP * 3; // [B, NP, COUT]

  char*  ws      = (char*)d_ws;
  int*   knn_idx = (int*)ws;                                          // B*NP*NS ints
  float* stats   = (float*)(ws + (size_t)B_ * NP * NS * sizeof(int)); // 1024 floats
  int*   fps_idx = (int*)(ws + (size_t)B_ * NP * NS * sizeof(int) + 1024 * sizeof(float));

  fps_kernel<<<B_, 512, 0, stream>>>(xyz, new_xyz, fps_idx);
  knn_kernel<<<(B_ * NP) / KNN_BLOCK, KNN_BLOCK, 0, stream>>>(xyz, new_xyz, knn_idx);
  zero_stats<<<1, 512, 0, stream>>>(stats);
  gemm_kernel<0><<<dim3(512, 2), 256, 0, stream>>>(xyz, features, W, gamma, beta,
                                                   new_xyz, knn_idx, stats, new_features);
  finalize_stats<<<1, 256, 0, stream>>>(stats);
  gemm_kernel<1><<<dim3(512, 2), 256, 0, stream>>>(xyz, features, W, gamma, beta,
                                                   new_xyz, knn_idx, stats, new_features);
}